// ID_Model_full_mean_24816321036423
// MI455X (gfx1250) — compile-verified
//
#include <hip/hip_runtime.h>

#define FEAT 384
#define QUADS (FEAT / 4) /* 96 threads per dst block = 3 full wave32 */

// ---------------- CSR construction ----------------

__global__ void zero_deg_kernel(int* __restrict__ deg, int n) {
    int i = blockIdx.x * blockDim.x + threadIdx.x;
    if (i < n) deg[i] = 0;
}

__global__ void count_kernel(const int* __restrict__ dst_idx,
                             int* __restrict__ deg, int num_edges) {
    int e = blockIdx.x * blockDim.x + threadIdx.x;
    if (e < num_edges) atomicAdd(&deg[dst_idx[e]], 1);
}

// Single-block exclusive scan of deg -> row_off (and cursor copy).
__global__ void scan_kernel(const int* __restrict__ deg,
                            int* __restrict__ row_off,
                            int* __restrict__ cursor, int n) {
    __shared__ int partial[1024];
    const int t = threadIdx.x;
    const int chunk = (n + 1023) >> 10;
    int base = t * chunk;
    int end = base + chunk;
    if (end > n) end = n;

    int s = 0;
    for (int i = base; i < end; ++i) s += deg[i];
    partial[t] = s;
    __syncthreads();

    // Hillis-Steele inclusive scan over 1024 partials.
    for (int off = 1; off < 1024; off <<= 1) {
        int v = partial[t];
        int add = (t >= off) ? partial[t - off] : 0;
        __syncthreads();
        partial[t] = v + add;
        __syncthreads();
    }

    int run = (t == 0) ? 0 : partial[t - 1];
    for (int i = base; i < end; ++i) {
        row_off[i] = run;
        cursor[i] = run;
        run += deg[i];
    }
    if (t == 1023) row_off[n] = partial[1023];
}

__global__ void scatter_kernel(const int* __restrict__ src_idx,
                               const int* __restrict__ dst_idx,
                               int* __restrict__ cursor,
                               int* __restrict__ csr_src, int num_edges) {
    int e = blockIdx.x * blockDim.x + threadIdx.x;
    if (e < num_edges) {
        int d = dst_idx[e];
        int pos = atomicAdd(&cursor[d], 1);
        csr_src[pos] = src_idx[e];
    }
}

// ---------------- Aggregation ----------------
// One 96-thread block per dst row. Neighbor rows are staged global->LDS with
// the CDNA5 async DMA path (global_load_async_to_lds_b128 / s_wait_asynccnt),
// double-buffered so the DMA of row e+1 overlaps accumulation of row e.

__device__ __forceinline__ void async_row_load(unsigned lds_addr,
                                               const float* gptr) {
    unsigned long long ga = (unsigned long long)(uintptr_t)gptr;
    asm volatile("global_load_async_to_lds_b128 %0, %1, off"
                 :: "v"(lds_addr), "v"(ga)
                 : "memory");
}

__global__ void __launch_bounds__(QUADS)
agg_kernel(const float* __restrict__ h,
           const int* __restrict__ row_off,
           const int* __restrict__ csr_src,
           float* __restrict__ out, int num_dst) {
    __shared__ float buf[2][FEAT];           // 2 x 1536 B row buffers
    const int d = blockIdx.x;
    const int t = threadIdx.x;               // 0..95: one float4 of the row
    const int beg = row_off[d];
    const int end = row_off[d + 1];

    // 32-bit LDS byte addresses of this thread's 16B slice in each buffer
    // (low 32 bits of the flat shared-aperture address == LDS offset).
    const unsigned lds0 = (unsigned)(uintptr_t)(&buf[0][t * 4]);
    const unsigned lds1 = (unsigned)(uintptr_t)(&buf[1][t * 4]);

    float4 acc = make_float4(0.f, 0.f, 0.f, 0.f);

    // Skip sources < num_dst: those rows are zeroed in the reference
    // (they still count toward the degree via row_off).
    int e = beg;
    while (e < end && csr_src[e] < num_dst) ++e;

    if (e < end) {
        async_row_load(lds0, h + (size_t)csr_src[e] * FEAT + t * 4);
        int cur = 0;
        while (e < end) {
            int en = e + 1;
            while (en < end && csr_src[en] < num_dst) ++en;
            if (en < end) {
                // Warm L2 for the DMA after next (global_prefetch_b8).
                if (en + 1 < end) {
                    int sp = csr_src[en + 1];
                    if (sp >= num_dst)
                        __builtin_prefetch(
                            (const void*)(h + (size_t)sp * FEAT + t * 4), 0, 0);
                }
                // Issue next row into the other buffer, then wait until only
                // that newest DMA is outstanding (ASYNCcnt is per-wave, in-order).
                async_row_load(cur ? lds0 : lds1,
                               h + (size_t)csr_src[en] * FEAT + t * 4);
                asm volatile("s_wait_asynccnt 0x1" ::: "memory");
            } else {
                asm volatile("s_wait_asynccnt 0x0" ::: "memory");
            }
            const float4* bp = (const float4*)(cur ? &buf[1][0] : &buf[0][0]);
            float4 v = bp[t];
            acc.x += v.x; acc.y += v.y; acc.z += v.z; acc.w += v.w;
            cur ^= 1;
            e = en;
        }
    }

    const int degv = end - beg;
    const float scale = (degv > 0) ? (1.0f / (float)degv) : 0.0f;

    float4 r;
    r.x = acc.x * scale;
    r.y = acc.y * scale;
    r.z = acc.z * scale;
    r.w = acc.w * scale;

    // out[d, 0:384]   = mean of neighbor messages
    ((float4*)(out + (size_t)d * (2 * FEAT)))[t] = r;
    // out[d, 384:768] = h[d]
    float4 hv = ((const float4*)(h + (size_t)d * FEAT))[t];
    ((float4*)(out + (size_t)d * (2 * FEAT) + FEAT))[t] = hv;
}

// ---------------- Launch ----------------

extern "C" void kernel_launch(void* const* d_in, const int* in_sizes, int n_in,
                              void* d_out, int out_size, void* d_ws, size_t ws_size,
                              hipStream_t stream) {
    const float* h       = (const float*)d_in[0];
    const int*   src_idx = (const int*)d_in[1];
    const int*   dst_idx = (const int*)d_in[2];
    // d_in[3] (num_dst scalar) lives on device; derive the value host-side:
    const int num_edges = in_sizes[1];
    const int num_dst   = out_size / (2 * FEAT);

    int* row_off = (int*)d_ws;                 // num_dst + 1
    int* cursor  = row_off + (num_dst + 1);    // num_dst
    int* deg     = cursor + num_dst;           // num_dst
    int* csr_src = deg + num_dst;              // num_edges
    float* out   = (float*)d_out;

    zero_deg_kernel<<<(num_dst + 255) / 256, 256, 0, stream>>>(deg, num_dst);
    count_kernel<<<(num_edges + 255) / 256, 256, 0, stream>>>(dst_idx, deg, num_edges);
    scan_kernel<<<1, 1024, 0, stream>>>(deg, row_off, cursor, num_dst);
    scatter_kernel<<<(num_edges + 255) / 256, 256, 0, stream>>>(src_idx, dst_idx,
                                                                cursor, csr_src, num_edges);
    agg_kernel<<<num_dst, QUADS, 0, stream>>>(h, row_off, csr_src, out, num_dst);
}